// voxelization_22522808500260
// MI455X (gfx1250) — compile-verified
//
#include <hip/hip_runtime.h>
#include <stdint.h>

// ============================================================================
// Voxelization + multi-scale sorted-unique for MI455X (gfx1250).
//
// Dense-bitmap unique (sorted order for free):
//   1) clear occupancy bitmap (<=16.7MB; L2-resident)
//   2) per point: quantize -> key (max 2^27, fits u32), NT-store full_coors,
//      atomicOr bit into bitmap
//   3) hierarchical exclusive scan of word popcounts (wave32 shfl + LDS)
//   4) inverse[i] = rank of point's bit (L2-resident random gathers)
//   5) scatter unique [b,z,y,x] rows in key-sorted order (NT b128 stores)
//   6) bump device-side running output offset by 5N + 4M
//
// L2 policy: points(32MB)+bitmap(16.7MB)+prefix(16.7MB) stay resident;
// all ~160MB of write-once outputs use non-temporal stores so they stream
// to HBM without evicting the reused working set.
// ============================================================================

#define TPB 256
#define NWAVE (TPB / 32)

typedef __attribute__((ext_vector_type(4))) float v4f;   // clang vector: NT-store OK

// ---- workspace layout (bytes); total ~42.1 MB ----
static constexpr size_t MAXN         = 2u * 1024u * 1024u;
static constexpr size_t MAX_WORDS    = 4194304;                 // 512*512*64*8/32
static constexpr size_t MAX_B0       = MAX_WORDS / 256;         // 16384

static constexpr size_t KEYS_OFF     = 0;
static constexpr size_t KEYS_BYTES   = MAXN * 4;
static constexpr size_t BITMAP_OFF   = KEYS_OFF + KEYS_BYTES;
static constexpr size_t BITMAP_BYTES = MAX_WORDS * 4;
static constexpr size_t PRE_OFF      = BITMAP_OFF + BITMAP_BYTES;
static constexpr size_t PRE_BYTES    = MAX_WORDS * 4;
static constexpr size_t BSUM_OFF     = PRE_OFF + PRE_BYTES;
static constexpr size_t BSUM_BYTES   = MAX_B0 * 4;
static constexpr size_t BSUMEX_OFF   = BSUM_OFF + BSUM_BYTES;
static constexpr size_t S2_OFF       = BSUMEX_OFF + BSUM_BYTES;
static constexpr size_t S2_BYTES     = 256 * 4;
static constexpr size_t S2EX_OFF     = S2_OFF + S2_BYTES;
static constexpr size_t SCAL_OFF     = S2EX_OFF + S2_BYTES;     // off:u64, M:u32

// ---------------------------------------------------------------------------
__global__ void k_init_off(unsigned long long* offp) { *offp = 0ull; }

__global__ void k_clear(uint4* bm4, int nQuads) {
    int i = blockIdx.x * TPB + threadIdx.x;
    if (i < nQuads) bm4[i] = make_uint4(0u, 0u, 0u, 0u);   // RT: prime L2 lines
}

// quantize + full_coors (NT) + key + bitmap atomicOr
__global__ void k_points(const float* __restrict__ pts,
                         const int* __restrict__ bidx,
                         unsigned* __restrict__ keys,
                         unsigned* __restrict__ bitmap,
                         float* __restrict__ out,
                         const unsigned long long* __restrict__ offp,
                         int n, int sx, int sy, int sz, int vec4ok) {
    int i = blockIdx.x * TPB + threadIdx.x;
    if (i >= n) return;
    __builtin_prefetch(pts + 3 * (i + 8 * TPB), 0, 0);   // global_prefetch_b8
    unsigned long long base = *offp;
    float px = pts[3 * i + 0];
    float py = pts[3 * i + 1];
    float pz = pts[3 * i + 2];
    int b = bidx[i];
    // match jnp: (dims * (p - lo)) / (hi - lo), truncate toward zero (>=0 here)
    int xi = (int)(((float)sx * (px + 51.2f)) / 102.4f);
    int yi = (int)(((float)sy * (py + 51.2f)) / 102.4f);
    int zi = (int)(((float)sz * (pz + 4.0f)) / 6.4f);
    unsigned key = ((unsigned)((b * sx + xi) * sy + yi)) * (unsigned)sz + (unsigned)zi;
    keys[i] = key;
    atomicOr(&bitmap[key >> 5], 1u << (key & 31u));
    float* fc = out + base + (size_t)i * 4;
    if (vec4ok) {   // uniform branch; 16B-aligned row -> one NT global_store_b128
        v4f row = { (float)b, (float)xi, (float)yi, (float)zi };
        __builtin_nontemporal_store(row, (v4f*)fc);
    } else {
        __builtin_nontemporal_store((float)b,  fc + 0);
        __builtin_nontemporal_store((float)xi, fc + 1);
        __builtin_nontemporal_store((float)yi, fc + 2);
        __builtin_nontemporal_store((float)zi, fc + 3);
    }
}

// ---- wave32 shuffle inclusive scan + LDS combine (2 barriers total) --------
__device__ __forceinline__ unsigned wave_scan_incl(unsigned v) {
    int lane = threadIdx.x & 31;
    #pragma unroll
    for (int o = 1; o < 32; o <<= 1) {
        unsigned x = __shfl_up(v, o, 32);
        if (lane >= o) v += x;
    }
    return v;
}

__device__ __forceinline__ unsigned block_scan_incl(unsigned v, unsigned* lds) {
    int t = threadIdx.x, lane = t & 31, wid = t >> 5;
    unsigned incl = wave_scan_incl(v);
    if (lane == 31) lds[wid] = incl;            // wave totals
    __syncthreads();
    if (wid == 0) {
        unsigned wv = (lane < NWAVE) ? lds[lane] : 0u;
        unsigned wincl = wave_scan_incl(wv);
        if (lane < NWAVE) lds[lane] = wincl - wv;   // exclusive wave offsets
    }
    __syncthreads();
    return incl + lds[wid];
}

// L0: per-word popcount scan; words staged into LDS via gfx1250 async DMA
// (GLOBAL_LOAD_ASYNC_TO_LDS_B32, tracked by ASYNCcnt).
__global__ void k_scan_words(const unsigned* __restrict__ bitmap,
                             unsigned* __restrict__ pre,
                             unsigned* __restrict__ bsum) {
    __shared__ unsigned stage[TPB];
    __shared__ unsigned sc[NWAVE];
    int t = threadIdx.x;
    int w = blockIdx.x * TPB + t;               // nWords is a multiple of 256
    unsigned lds_addr = (unsigned)(size_t)&stage[t];   // low 32 bits = LDS offset
    unsigned long long gaddr = (unsigned long long)(size_t)(bitmap + w);
    asm volatile("global_load_async_to_lds_b32 %0, %1, off"
                 :: "v"(lds_addr), "v"(gaddr) : "memory");
    asm volatile("s_wait_asynccnt 0" ::: "memory");
    __syncthreads();
    unsigned c = __popc(stage[t]);
    unsigned incl = block_scan_incl(c, sc);
    pre[w] = incl - c;                          // block-local exclusive prefix
    if (t == TPB - 1) bsum[blockIdx.x] = incl;
}

// L1: scan of block sums
__global__ void k_scan_vals(const unsigned* __restrict__ vals,
                            unsigned* __restrict__ valsEx,
                            unsigned* __restrict__ s2, int nv) {
    __shared__ unsigned sc[NWAVE];
    int t = threadIdx.x;
    int i = blockIdx.x * TPB + t;
    unsigned v = (i < nv) ? vals[i] : 0u;
    unsigned incl = block_scan_incl(v, sc);
    if (i < nv) valsEx[i] = incl - v;
    if (t == TPB - 1) s2[blockIdx.x] = incl;
}

// L2: single-block scan of <=256 level-2 sums; total -> M
__global__ void k_scan_small(const unsigned* __restrict__ s2,
                             unsigned* __restrict__ s2Ex,
                             unsigned* __restrict__ Mp, int nv) {
    __shared__ unsigned sc[NWAVE];
    int t = threadIdx.x;
    unsigned v = (t < nv) ? s2[t] : 0u;
    unsigned incl = block_scan_incl(v, sc);
    if (t < nv) s2Ex[t] = incl - v;
    if (t == TPB - 1) *Mp = incl;               // total unique count M
}

__device__ __forceinline__ unsigned word_rank(unsigned w,
                                              const unsigned* pre,
                                              const unsigned* bEx,
                                              const unsigned* s2Ex) {
    return pre[w] + bEx[w >> 8] + s2Ex[w >> 16];
}

// inverse[i] = global rank of this point's bit (L2-resident gathers, NT out)
__global__ void k_inverse(const unsigned* __restrict__ keys,
                          const unsigned* __restrict__ bitmap,
                          const unsigned* __restrict__ pre,
                          const unsigned* __restrict__ bEx,
                          const unsigned* __restrict__ s2Ex,
                          float* __restrict__ out,
                          const unsigned long long* __restrict__ offp, int n) {
    int i = blockIdx.x * TPB + threadIdx.x;
    if (i >= n) return;
    unsigned key = keys[i];
    unsigned w = key >> 5;
    unsigned mask = (1u << (key & 31u)) - 1u;
    unsigned rank = word_rank(w, pre, bEx, s2Ex) + __popc(bitmap[w] & mask);
    __builtin_nontemporal_store((float)rank,
        out + *offp + 4ull * (unsigned long long)n + (unsigned long long)i);
}

// scatter unique [b,z,y,x] rows in key-sorted order (NT b128 stores)
__global__ void k_scatter(const unsigned* __restrict__ bitmap,
                          const unsigned* __restrict__ pre,
                          const unsigned* __restrict__ bEx,
                          const unsigned* __restrict__ s2Ex,
                          float* __restrict__ out,
                          const unsigned long long* __restrict__ offp,
                          int nWords, int n, int sx, int sy, int sz, int vec4ok) {
    int w = blockIdx.x * TPB + threadIdx.x;
    if (w >= nWords) return;
    unsigned bits = bitmap[w];
    if (!bits) return;
    unsigned base = word_rank((unsigned)w, pre, bEx, s2Ex);
    unsigned long long cbase = *offp + 5ull * (unsigned long long)n;
    unsigned cnt = 0;
    while (bits) {
        int j = __ffs(bits) - 1;
        unsigned key = ((unsigned)w << 5) | (unsigned)j;
        unsigned zb = key % (unsigned)sz; unsigned rem = key / (unsigned)sz;
        unsigned yb = rem % (unsigned)sy; rem /= (unsigned)sy;
        unsigned xb = rem % (unsigned)sx; unsigned bb = rem / (unsigned)sx;
        float* c = out + cbase + (size_t)(base + cnt) * 4;
        if (vec4ok) {
            v4f row = { (float)bb, (float)zb, (float)yb, (float)xb };
            __builtin_nontemporal_store(row, (v4f*)c);
        } else {
            __builtin_nontemporal_store((float)bb, c + 0);
            __builtin_nontemporal_store((float)zb, c + 1);
            __builtin_nontemporal_store((float)yb, c + 2);
            __builtin_nontemporal_store((float)xb, c + 3);
        }
        bits &= bits - 1u; ++cnt;
    }
}

__global__ void k_bump(unsigned long long* offp, const unsigned* Mp, int n) {
    *offp += 5ull * (unsigned long long)n + 4ull * (unsigned long long)(*Mp);
}

// ---------------------------------------------------------------------------
extern "C" void kernel_launch(void* const* d_in, const int* in_sizes, int n_in,
                              void* d_out, int out_size, void* d_ws, size_t ws_size,
                              hipStream_t stream) {
    const float* pts  = (const float*)d_in[0];
    const int*   bidx = (const int*)d_in[1];
    float* out = (float*)d_out;
    char*  ws  = (char*)d_ws;

    unsigned* keys   = (unsigned*)(ws + KEYS_OFF);
    unsigned* bitmap = (unsigned*)(ws + BITMAP_OFF);
    unsigned* pre    = (unsigned*)(ws + PRE_OFF);
    unsigned* bsum   = (unsigned*)(ws + BSUM_OFF);
    unsigned* bsumEx = (unsigned*)(ws + BSUMEX_OFF);
    unsigned* s2     = (unsigned*)(ws + S2_OFF);
    unsigned* s2Ex   = (unsigned*)(ws + S2EX_OFF);
    unsigned long long* offp = (unsigned long long*)(ws + SCAL_OFF);
    unsigned* Mp = (unsigned*)(ws + SCAL_OFF + 8);

    const int n = in_sizes[1];                 // N points
    const int nPtBlk = (n + TPB - 1) / TPB;
    // 16B alignment of every output row holds iff N % 4 == 0 (5N, 4M ≡ 0 mod 4)
    const int vec4ok = ((n & 3) == 0) ? 1 : 0;

    k_init_off<<<1, 1, 0, stream>>>(offp);

    const int scales[4] = {2, 4, 8, 1};
    for (int s = 0; s < 4; ++s) {
        const int sc = scales[s];
        const int sx = (512 + sc - 1) / sc;
        const int sy = (512 + sc - 1) / sc;
        const int sz = (64 + sc - 1) / sc;
        const int nWords = (8 * sx * sy * sz) / 32;   // always multiple of 256
        const int nB0 = nWords / TPB;
        const int nB1 = (nB0 + TPB - 1) / TPB;        // <= 64

        k_clear<<<(nWords / 4 + TPB - 1) / TPB, TPB, 0, stream>>>((uint4*)bitmap, nWords / 4);
        k_points<<<nPtBlk, TPB, 0, stream>>>(pts, bidx, keys, bitmap, out, offp,
                                             n, sx, sy, sz, vec4ok);
        k_scan_words<<<nB0, TPB, 0, stream>>>(bitmap, pre, bsum);
        k_scan_vals<<<nB1, TPB, 0, stream>>>(bsum, bsumEx, s2, nB0);
        k_scan_small<<<1, TPB, 0, stream>>>(s2, s2Ex, Mp, nB1);
        k_inverse<<<nPtBlk, TPB, 0, stream>>>(keys, bitmap, pre, bsumEx, s2Ex, out, offp, n);
        k_scatter<<<nB0, TPB, 0, stream>>>(bitmap, pre, bsumEx, s2Ex, out, offp,
                                           nWords, n, sx, sy, sz, vec4ok);
        k_bump<<<1, 1, 0, stream>>>(offp, Mp, n);
    }
}